// NECTAR_Binning_79070347919529
// MI455X (gfx1250) — compile-verified
//
#include <hip/hip_runtime.h>
#include <stdint.h>

// Problem constants (match reference)
#define B_    16
#define C_    8
#define H_    512
#define W_    512
#define HW_   (H_ * W_)
#define NBINS 15

// Tiling
#define TW 32
#define TH 32
#define HALO_W (TW + 2)          // 34
#define HALO_H (TH + 2)          // 34
#define PRED_STRIDE 36           // padded row stride for the u8 pred tile

typedef __attribute__((ext_vector_type(4))) unsigned int u32x4;
typedef __attribute__((ext_vector_type(8))) int          i32x8;
typedef __attribute__((ext_vector_type(4))) int          i32x4;

__device__ __forceinline__ int iclamp(int v, int lo, int hi) {
    return v < lo ? lo : (v > hi ? hi : v);
}

__global__ __launch_bounds__(256)
void NECTAR_Binning_79070347919529_kernel(const float* __restrict__ logits,
                                          const float* __restrict__ val_freqs,
                                          float* __restrict__ out)
{
    __shared__ float         s_vf[C_ * 9 * NBINS];            // 1080 floats, 4320 B
    __shared__ unsigned char s_pred[HALO_H * PRED_STRIDE];    // pred tile + halo

    const int tid = threadIdx.x;
    const int b   = blockIdx.z;
    const int h0  = blockIdx.y * TH;
    const int w0  = blockIdx.x * TW;

    const size_t img_base = (size_t)b * (size_t)(C_ * HW_);

    // ---------------------------------------------------------------
    // TDM: async DMA of the calibration table (1080 f32) into LDS.
    // Descriptor per CDNA5 ISA ch.8 (D# group0/group1), issued by wave 0.
    // ---------------------------------------------------------------
    if (tid < 32) {
        // low 32 bits of a generic LDS address == LDS byte offset (ISA 10.2)
        unsigned lds_off = (unsigned)(uintptr_t)(void*)&s_vf[0];
        uint64_t ga = (uint64_t)(uintptr_t)val_freqs;

        u32x4 g0;
        g0[0] = 1u;                                            // count=1, user desc
        g0[1] = lds_off;                                       // lds_addr (bytes)
        g0[2] = (unsigned)(ga & 0xFFFFFFFFu);                  // global_addr[31:0]
        g0[3] = (unsigned)((ga >> 32) & 0x1FFFFFFu)            // global_addr[56:32]
              | (2u << 30);                                    // type=2 ("image")

        i32x8 g1;
        g1[0] = (int)(2u << 16);        // wg_mask=0 (not in cluster), data_size=4B
        g1[1] = (int)(1080u << 16);     // tensor_dim0 lo16 (1080 elements)
        g1[2] = (int)(1u << 16);        // tensor_dim0 hi=0 | tensor_dim1 lo16 = 1
        g1[3] = (int)(1080u << 16);     // tensor_dim1 hi=0 | tile_dim0 = 1080
        g1[4] = 1;                      // tile_dim1 = 1, tile_dim2 = 0 (unused)
        g1[5] = 1080;                   // tensor_dim0_stride lo32
        g1[6] = (int)(1080u << 16);     // stride0 hi=0 | tensor_dim1_stride lo16
        g1[7] = 0;                      // tensor_dim1_stride hi

        i32x4 gz4 = {0, 0, 0, 0};                  // groups 2/3: dims >2D unused
        i32x8 gz8 = {0, 0, 0, 0, 0, 0, 0, 0};      // extra group (clang-23 6-arg form)
        __builtin_amdgcn_tensor_load_to_lds(g0, g1, gz4, gz4, gz8, /*cpol=*/0);
    }

    // ---------------------------------------------------------------
    // Phase A(a): own 4 pixels -> logits in registers, argmax -> LDS pred
    // ---------------------------------------------------------------
    const int tx = tid & (TW - 1);
    const int ty = tid >> 5;            // 0..7
    float lg[4][C_];
    int   pred[4];

#pragma unroll
    for (int k = 0; k < 4; ++k) {
        const int r = ty + 8 * k;
        const size_t rb = img_base + (size_t)(h0 + r) * W_ + (size_t)(w0 + tx);
        float bestv = -3.0e38f; int besti = 0;
#pragma unroll
        for (int c = 0; c < C_; ++c) {
            float v = logits[rb + (size_t)c * HW_];
            lg[k][c] = v;
            if (v > bestv) { bestv = v; besti = c; }
        }
        pred[k] = besti;
        s_pred[(r + 1) * PRED_STRIDE + (tx + 1)] = (unsigned char)besti;
    }

    // ---------------------------------------------------------------
    // Phase A(b): 132 halo-ring preds (sentinel class 8 outside the image)
    // ---------------------------------------------------------------
    if (tid < 132) {
        int i, j;
        if      (tid < 34)  { i = 0;             j = tid;        }
        else if (tid < 68)  { i = HALO_H - 1;    j = tid - 34;   }
        else if (tid < 100) { i = tid - 68 + 1;  j = 0;          }
        else                { i = tid - 100 + 1; j = HALO_W - 1; }
        const int gh = h0 - 1 + i;
        const int gw = w0 - 1 + j;
        int p = 8;                                   // OOB sentinel (never matches)
        if (gh >= 0 && gh < H_ && gw >= 0 && gw < W_) {
            const size_t rb = img_base + (size_t)gh * W_ + (size_t)gw;
            float bestv = -3.0e38f; int besti = 0;
#pragma unroll
            for (int c = 0; c < C_; ++c) {
                float v = logits[rb + (size_t)c * HW_];
                if (v > bestv) { bestv = v; besti = c; }
            }
            p = besti;
        }
        s_pred[i * PRED_STRIDE + j] = (unsigned char)p;
    }

    if (tid < 32) __builtin_amdgcn_s_wait_tensorcnt(0);   // wave0: TDM done
    __syncthreads();                                      // publish s_vf + s_pred

    // ---------------------------------------------------------------
    // Phase B: softmax + neighbor histogram + bin + gather + normalize
    // ---------------------------------------------------------------
#pragma unroll
    for (int k = 0; k < 4; ++k) {
        const int r = ty + 8 * k;

        // nibble-packed class histogram over the 3x3 window (center incl.)
        const unsigned char* row0 = &s_pred[(r    ) * PRED_STRIDE + tx];
        const unsigned char* row1 = &s_pred[(r + 1) * PRED_STRIDE + tx];
        const unsigned char* row2 = &s_pred[(r + 2) * PRED_STRIDE + tx];
        uint64_t acc = (uint64_t)1 << (4 * pred[k]);           // center from reg
        acc += (uint64_t)1 << (4 * (int)row0[0]);
        acc += (uint64_t)1 << (4 * (int)row0[1]);
        acc += (uint64_t)1 << (4 * (int)row0[2]);
        acc += (uint64_t)1 << (4 * (int)row1[0]);
        acc += (uint64_t)1 << (4 * (int)row1[2]);
        acc += (uint64_t)1 << (4 * (int)row2[0]);
        acc += (uint64_t)1 << (4 * (int)row2[1]);
        acc += (uint64_t)1 << (4 * (int)row2[2]);
        const unsigned accLo = (unsigned)acc;                  // classes 0..7 only
        const int n_inb = (int)((accLo * 0x11111111u) >> 28);  // in-bounds window size

        // softmax over C=8 (registers only)
        float m = lg[k][0];
#pragma unroll
        for (int c = 1; c < C_; ++c) m = fmaxf(m, lg[k][c]);
        float e[C_]; float s = 0.f;
#pragma unroll
        for (int c = 0; c < C_; ++c) { e[c] = __expf(lg[k][c] - m); s += e[c]; }
        const float inv = __builtin_amdgcn_rcpf(s);

        float cal[C_]; float csum = 0.f;
        const int ctr = pred[k];
#pragma unroll
        for (int c = 0; c < C_; ++c) {
            const int wc  = (int)((accLo >> (4 * c)) & 0xFu);  // window count of class c
            int cnt = (c == ctr) ? (wc - 1) : (n_inb - wc - 1);
            cnt = iclamp(cnt, 0, 8);
            const float p = e[c] * inv;
            int bin = (int)(p * (float)NBINS);
            bin = bin > (NBINS - 1) ? (NBINS - 1) : bin;       // p>=0 -> bin>=0
            const float v = s_vf[c * (9 * NBINS) + cnt * NBINS + bin];
            cal[c] = v; csum += v;
        }
        const float rinv = __builtin_amdgcn_rcpf(csum);

        const size_t ob = img_base + (size_t)(h0 + r) * W_ + (size_t)(w0 + tx);
#pragma unroll
        for (int c = 0; c < C_; ++c)
            __builtin_nontemporal_store(cal[c] * rinv, &out[ob + (size_t)c * HW_]);
    }
}

extern "C" void kernel_launch(void* const* d_in, const int* in_sizes, int n_in,
                              void* d_out, int out_size, void* d_ws, size_t ws_size,
                              hipStream_t stream) {
    const float* logits    = (const float*)d_in[0];   // [16,8,512,512] f32
    const float* val_freqs = (const float*)d_in[1];   // [8,9,15] f32
    float* out             = (float*)d_out;           // [16,8,512,512] f32
    (void)in_sizes; (void)n_in; (void)out_size; (void)d_ws; (void)ws_size;

    dim3 grid(W_ / TW, H_ / TH, B_);                  // (16,16,16)
    NECTAR_Binning_79070347919529_kernel<<<grid, 256, 0, stream>>>(logits, val_freqs, out);
}